// GraphSageLayer_83382495084581
// MI455X (gfx1250) — compile-verified
//
#include <hip/hip_runtime.h>

// Problem constants (fixed by the reference)
#define NB   10000          // nodes
#define FB   128            // feature dim (in == out)
#define EB   160000         // edges
#define BTC  4              // B*T
#define LN   2              // layers
#define ROWS (NB*BTC)       // 40000 rows of the flattened [N*BT, F] matrix
#define MT   (ROWS/16)      // 2500 M-tiles of 16 rows

typedef __attribute__((ext_vector_type(2))) float v2f;
typedef __attribute__((ext_vector_type(8))) float v8f;

// ---------------- utility kernels ----------------

__global__ void k_zero_f(float* __restrict__ p, int n) {
    int i = blockIdx.x * blockDim.x + threadIdx.x;
    if (i < n) p[i] = 0.0f;
}

__global__ void k_zero_f4(float4* __restrict__ p, int n4) {
    int i = blockIdx.x * blockDim.x + threadIdx.x;
    if (i < n4) p[i] = make_float4(0.f, 0.f, 0.f, 0.f);
}

// feature [BT, N, F] -> h [N, BT, F]   (float4 moves)
__global__ void k_transpose_in(const float* __restrict__ feat, float* __restrict__ h) {
    int i = blockIdx.x * blockDim.x + threadIdx.x;   // over BT*N*(F/4)
    if (i >= BTC * NB * (FB / 4)) return;
    int f4 = i & 31;                 // F/4 = 32
    int n  = (i >> 5) % NB;
    int bt = i / (32 * NB);
    float4 v = ((const float4*)feat)[(bt * NB + n) * 32 + f4];
    ((float4*)h)[(n * BTC + bt) * 32 + f4] = v;
}

__global__ void k_deg(const int* __restrict__ dst, float* __restrict__ deg) {
    int e = blockIdx.x * blockDim.x + threadIdx.x;
    if (e < EB) unsafeAtomicAdd(&deg[dst[e]], 1.0f);   // exact (integer sums < 2^24)
}

__global__ void k_invdeg(float* __restrict__ deg) {
    int i = blockIdx.x * blockDim.x + threadIdx.x;
    if (i < NB) {
        float d = deg[i];
        deg[i] = (d > 0.0f) ? (1.0f / d) : 0.0f;
    }
}

// Repack W_self/W_neigh [L,K=128,Ncol=128] into interleaved B-fragment layout:
// Wsn[l][(k>>1)*128 + col] (float4) = { Ws[k,col], Ws[k+1,col], Wn[k,col], Wn[k+1,col] }
// -> one global_load_b128 per column-tile per K-step in the GEMM.
__global__ void k_repack_w(const float* __restrict__ Ws, const float* __restrict__ Wn,
                           float* __restrict__ Wsn) {
    int i = blockIdx.x * blockDim.x + threadIdx.x;   // over LN*FB*FB = 32768
    if (i >= LN * FB * FB) return;
    int col = i & (FB - 1);
    int k   = (i >> 7) & (FB - 1);
    int l   = i >> 14;
    size_t src = (size_t)l * FB * FB + (size_t)k * FB + col;
    size_t dst = ((size_t)l * (FB / 2) * FB + (size_t)(k >> 1) * FB + col) * 4 + (k & 1);
    Wsn[dst]     = Ws[src];
    Wsn[dst + 2] = Wn[src];
}

// gather h[src] rows, scatter-add into agg[dst] rows.
// 128 consecutive threads per edge; 32 consecutive threads read one
// contiguous 512B row (coalesced float4); L2-resident atomics.
__global__ void k_scatter(const float* __restrict__ h, const int* __restrict__ src,
                          const int* __restrict__ dst, float* __restrict__ agg) {
    int i = blockIdx.x * blockDim.x + threadIdx.x;   // total EB*BT*32 = 20.48M
    if (i >= EB * BTC * 32) return;
    int f4 = i & 31;
    int c  = (i >> 5) & 3;
    int e  = i >> 7;
    int s = src[e];
    int d = dst[e];
    float4 v = ((const float4*)h)[(s * BTC + c) * 32 + f4];
    float* ap = agg + ((size_t)(d * BTC + c) * FB + f4 * 4);
    unsafeAtomicAdd(ap + 0, v.x);
    unsafeAtomicAdd(ap + 1, v.y);
    unsafeAtomicAdd(ap + 2, v.z);
    unsafeAtomicAdd(ap + 3, v.w);
}

// ---------------- fused WMMA GEMM ----------------
// Out[r, :] = A[r,:] @ Wself + (Agg[r,:] * invdeg[r/BT]) @ Wneigh + bias
// One wave computes a 16x128 output stripe: 8 f32 accumulators,
// 512 v_wmma_f32_16x16x4_f32, 10 VMEM issues per 16 WMMA in the K-loop.
__global__ void __launch_bounds__(256)
k_gemm(const float* __restrict__ A, const float* __restrict__ Agg,
       const float* __restrict__ invdeg, const float4* __restrict__ Wp,
       const float* __restrict__ bias, float* __restrict__ Out, int transposed) {
    int wave = blockIdx.x * (blockDim.x >> 5) + (threadIdx.x >> 5);
    if (wave >= MT) return;                       // wave-uniform, EXEC stays all-ones
    int lane = threadIdx.x & 31;
    int half = lane >> 4;                         // 0: K pair {0,1}/M 0..7; 1: K pair {2,3}/M 8..15
    int l16  = lane & 15;
    int row0 = wave * 16;

    // C layout: elem(v, lane) = C[M = v + 8*half, N = l16]; bias depends on N only.
    v8f acc[8];
#pragma unroll
    for (int j = 0; j < 8; ++j) {
        float bv = bias[j * 16 + l16];
        acc[j] = (v8f){bv, bv, bv, bv, bv, bv, bv, bv};
    }

    const int arow = row0 + l16;                  // A-frag: lane holds row l16, K = k0+2*half {,+1}
    const float idg = invdeg[arow >> 2];          // node id = row / BT

    for (int k0 = 0; k0 < FB; k0 += 4) {
        const int ka = k0 + half * 2;
        const int kp = (k0 >> 1) + half;          // K-pair index into repacked W
        v2f a_s, a_n;
        a_s.x = A[(size_t)arow * FB + ka];
        a_s.y = A[(size_t)arow * FB + ka + 1];
        a_n.x = Agg[(size_t)arow * FB + ka] * idg;     // fold mean-scaling into A rows
        a_n.y = Agg[(size_t)arow * FB + ka + 1] * idg;
#pragma unroll
        for (int j = 0; j < 8; ++j) {
            const int col = j * 16 + l16;
            float4 bb = Wp[(size_t)kp * FB + col];     // b128: both matrices' fragments
            v2f b_s, b_n;
            b_s.x = bb.x; b_s.y = bb.y;
            b_n.x = bb.z; b_n.y = bb.w;
            acc[j] = __builtin_amdgcn_wmma_f32_16x16x4_f32(
                false, a_s, false, b_s, (short)0, acc[j], false, false);
            acc[j] = __builtin_amdgcn_wmma_f32_16x16x4_f32(
                false, a_n, false, b_n, (short)0, acc[j], false, false);
        }
    }

#pragma unroll
    for (int j = 0; j < 8; ++j) {
        const int col = j * 16 + l16;
#pragma unroll
        for (int v = 0; v < 8; ++v) {
            int r = row0 + v + half * 8;
            size_t oidx;
            if (transposed) {                      // final layer writes [B,T,N,F] directly
                int n = r >> 2, bt = r & 3;
                oidx = ((size_t)bt * NB + n) * FB + col;
            } else {
                oidx = (size_t)r * FB + col;
            }
            Out[oidx] = acc[j][v];
        }
    }
}

// ---------------- host driver ----------------

extern "C" void kernel_launch(void* const* d_in, const int* in_sizes, int n_in,
                              void* d_out, int out_size, void* d_ws, size_t ws_size,
                              hipStream_t stream) {
    const float* feat   = (const float*)d_in[0];   // [B,T,N,F]
    const float* Wself  = (const float*)d_in[1];   // [L,F,F]
    const float* Wneigh = (const float*)d_in[2];   // [L,F,F]
    const float* bias   = (const float*)d_in[3];   // [L,F]
    const int*   esrc   = (const int*)d_in[4];     // [E]
    const int*   edst   = (const int*)d_in[5];     // [E]
    float* out = (float*)d_out;

    float* ws  = (float*)d_ws;
    float* h0  = ws;                                // [ROWS, F]
    float* h1  = h0 + (size_t)ROWS * FB;            // [ROWS, F]
    float* agg = h1 + (size_t)ROWS * FB;            // [ROWS, F]
    float* idg = agg + (size_t)ROWS * FB;           // [N]
    float* wsn = idg + (size_t)NB;                  // [L, F/2, F, 4] repacked weights

    // 0) repack weights into b128 fragment layout
    {
        int total = LN * FB * FB;
        k_repack_w<<<(total + 255) / 256, 256, 0, stream>>>(Wself, Wneigh, wsn);
    }
    // 1) layout transform to node-major
    {
        int total = BTC * NB * (FB / 4);
        k_transpose_in<<<(total + 255) / 256, 256, 0, stream>>>(feat, h0);
    }
    // 2) in-degree -> inverse degree
    k_zero_f<<<(NB + 255) / 256, 256, 0, stream>>>(idg, NB);
    k_deg<<<(EB + 255) / 256, 256, 0, stream>>>(edst, idg);
    k_invdeg<<<(NB + 255) / 256, 256, 0, stream>>>(idg);

    // 3) layers
    const float* hin = h0;
    float* hspare = h1;
    for (int l = 0; l < LN; ++l) {
        int n4 = ROWS * (FB / 4);
        k_zero_f4<<<(n4 + 255) / 256, 256, 0, stream>>>((float4*)agg, n4);

        int sthreads = EB * BTC * 32;
        k_scatter<<<(sthreads + 255) / 256, 256, 0, stream>>>(hin, esrc, edst, agg);

        bool last = (l == LN - 1);
        float* hout = last ? out : hspare;
        int blocks = (MT + 7) / 8;                 // 8 waves/block, 1 M-tile/wave
        k_gemm<<<blocks, 256, 0, stream>>>(hin, agg, idg,
                                           (const float4*)(wsn + (size_t)l * (FB / 2) * FB * 4),
                                           bias + (size_t)l * FB,
                                           hout, last ? 1 : 0);
        if (!last) {
            const float* t = hin;
            hin = hout;
            hspare = (float*)t;
        }
    }
}